// QNN_103079215597
// MI455X (gfx1250) — compile-verified
//
#include <hip/hip_runtime.h>

// CDNA5 / gfx1250: wave32, WMMA f32 16x16x4.
typedef __attribute__((ext_vector_type(2))) float v2f;
typedef __attribute__((ext_vector_type(8))) float v8f;

#define QNN_DIM 256   // 2^8 amplitudes
#define WAVES_PER_BLOCK 8

// Per-wave LDS ordering: DS ops from one wave complete in order; this waits for
// outstanding DS ops and stops the compiler from reordering around it.
__device__ __forceinline__ void lds_fence() {
  asm volatile("s_wait_dscnt 0" ::: "memory");
}

// Entry [m,k] of R(t0) (x) R(t1) (x) R(t2) (x) R(t3), with R = [[c,-s],[s,c]].
// Qubit i corresponds to bit (3-i) of the 4-bit index (qubit 0 = MSB).
__device__ __forceinline__ float kron4(const float* c, const float* s, int m, int k) {
  float p = 1.0f;
#pragma unroll
  for (int i = 0; i < 4; ++i) {
    int mb = (m >> (3 - i)) & 1;
    int kb = (k >> (3 - i)) & 1;
    float f = (mb == kb) ? c[i] : (mb ? s[i] : -s[i]);
    p *= f;
  }
  return p;
}

__global__ __launch_bounds__(32 * WAVES_PER_BLOCK) void qnn_wmma_kernel(
    const float* __restrict__ x,    // (B, 16)
    const float* __restrict__ w,    // (2, 8)
    const float* __restrict__ sc,   // (2, 8)
    float* __restrict__ out,        // (B, 32)
    int B) {
  __shared__ float smem[WAVES_PER_BLOCK * QNN_DIM];

  const int lane = threadIdx.x & 31;
  const int wv   = threadIdx.x >> 5;
  const int b    = blockIdx.x * WAVES_PER_BLOCK + wv;
  if (b >= B) return;                 // wave-uniform: EXEC stays all-ones
  float* sm = smem + wv * QNN_DIM;

  // ---- angles (wave-uniform; hardware v_sin_f32 / v_cos_f32) ---------------
  const float* xb = x + b * 16;
  float c1[8], s1[8], c2[8], s2[8];
#pragma unroll
  for (int i = 0; i < 8; ++i) {       // layer 0: features 0..7
    float t = 0.5f * __builtin_fmaf(sc[i], xb[i], w[i]);
    s1[i] = __sinf(t);
    c1[i] = __cosf(t);
  }
#pragma unroll
  for (int i = 0; i < 8; ++i) {       // layer 1: features 8..15
    float t = 0.5f * __builtin_fmaf(sc[8 + i], xb[8 + i], w[8 + i]);
    s2[i] = __sinf(t);
    c2[i] = __cosf(t);
  }

  const int h  = lane >> 4;           // wave half
  const int n  = lane & 15;           // column (B/C layout) / M index (A layout)
  const int n3 = (n >> 3) & 1;

  // ---- B operand = state after layer-1 RYs + CNOT cascade, analytically ----
  // amp[src] = prod_i (bit_i(src) ? s1[i] : c1[i]);  src = t ^ (t>>1), t = 16k+n.
  // src bits: b7=k3, b6=k3^k2, b5=k2^h, b4=h^p, b3=n3^p, b2=n3^n2, b1=n2^n1, b0=n1^n0
  // (k = 4*kk + p + 2*h), so the 8 entries per lane factor as qk[kk] * qp[p].
  int b2 = ((n >> 2) ^ (n >> 3)) & 1;
  int b1 = ((n >> 1) ^ (n >> 2)) & 1;
  int b0 = (n ^ (n >> 1)) & 1;
  float Pn = (b2 ? s1[5] : c1[5]) * (b1 ? s1[6] : c1[6]) * (b0 ? s1[7] : c1[7]);

  float qp[2];
#pragma unroll
  for (int p = 0; p < 2; ++p) {
    int bb4 = h ^ p;
    int bb3 = n3 ^ p;
    qp[p] = (bb4 ? s1[3] : c1[3]) * (bb3 ? s1[4] : c1[4]) * Pn;
  }
  float qk[4];
#pragma unroll
  for (int kk = 0; kk < 4; ++kk) {
    int k3 = (kk >> 1) & 1, k2 = kk & 1;    // compile-time after unroll
    int bb7 = k3, bb6 = k3 ^ k2;
    int bb5 = k2 ^ h;
    qk[kk] = (bb7 ? s1[0] : c1[0]) * (bb6 ? s1[1] : c1[1]) * (bb5 ? s1[2] : c1[2]);
  }

  v2f aop[4], bop[4];
#pragma unroll
  for (int kk = 0; kk < 4; ++kk) {
#pragma unroll
    for (int p = 0; p < 2; ++p) {
      int k = 4 * kk + p + 2 * h;
      bop[kk][p] = qk[kk] * qp[p];          // S'[k,n] in B layout, no LDS needed
      aop[kk][p] = kron4(c2, s2, n, k);     // L2[m=lane,k] generated in A layout
    }
  }

  // ---- D = L2 * S' ----------------------------------------------------------
  v8f acc = {};
#pragma unroll
  for (int kk = 0; kk < 4; ++kk)
    acc = __builtin_amdgcn_wmma_f32_16x16x4_f32(false, aop[kk], false, bop[kk],
                                                (short)0, acc, false, false);

  // ---- re-layout D through LDS (C layout -> A layout); F = D * R2^T --------
#pragma unroll
  for (int v = 0; v < 8; ++v) sm[16 * (v + 8 * h) + n] = acc[v];
  lds_fence();
#pragma unroll
  for (int kk = 0; kk < 4; ++kk) {
#pragma unroll
    for (int p = 0; p < 2; ++p) {
      int k = 4 * kk + p + 2 * h;
      aop[kk][p] = sm[16 * n + k];              // A = D: lane holds row m = n
      bop[kk][p] = kron4(c2 + 4, s2 + 4, n, k); // B[k,n] = R2[n,k] (i.e. R2^T)
    }
  }
  v8f acc2 = {};
#pragma unroll
  for (int kk = 0; kk < 4; ++kk)
    acc2 = __builtin_amdgcn_wmma_f32_16x16x4_f32(false, aop[kk], false, bop[kk],
                                                 (short)0, acc2, false, false);

  // ---- CNOT cascade #2 + measurement ---------------------------------------
#pragma unroll
  for (int v = 0; v < 8; ++v) sm[16 * (v + 8 * h) + n] = acc2[v];
  lds_fence();

  // probs[b, g=lane] = sum_t F[gray(8g+t)]^2.  Flipping t bit0 flips only
  // src bit0, so the 8 Gray-scattered reads pair into 4 aligned b64 loads.
  float pr = 0.0f;
#pragma unroll
  for (int u = 0; u < 4; ++u) {
    int idx  = 8 * lane + 2 * u;
    int src  = idx ^ (idx >> 1);
    int base = src & ~1;
    v2f vv = *(const v2f*)(sm + base);
    pr += vv[0] * vv[0] + vv[1] * vv[1];
  }
  out[b * 32 + lane] = pr;
}

extern "C" void kernel_launch(void* const* d_in, const int* in_sizes, int n_in,
                              void* d_out, int out_size, void* d_ws, size_t ws_size,
                              hipStream_t stream) {
  const float* x  = (const float*)d_in[0];   // inputs   (B,16) f32
  const float* w  = (const float*)d_in[1];   // weights  (2,8)  f32
  const float* sc = (const float*)d_in[2];   // scalings (2,8)  f32
  float* out = (float*)d_out;                // probs (B,32) f32

  const int B = in_sizes[0] / 16;
  const int blocks = (B + WAVES_PER_BLOCK - 1) / WAVES_PER_BLOCK;
  qnn_wmma_kernel<<<blocks, 32 * WAVES_PER_BLOCK, 0, stream>>>(x, w, sc, out, B);
}